// CDF_68444598829585
// MI455X (gfx1250) — compile-verified
//
#include <hip/hip_runtime.h>
#include <math.h>

// ---------------------------------------------------------------------------
// MI455X (gfx1250) per-channel 1->3->3->3->1 MLP, ch=256, N=65536.
// Memory-bound: 64MB in + 64MB out => ~5.5us floor @ 23.3 TB/s.
// CDNA5 paths: clause of 4 global_load_async_to_lds_b128 (ASYNCcnt, NT hint,
// shared VGPR pair + IOFFSET on both LDS and global sides), descending
// s_wait_asynccnt drain, native v_tanh_f32 (TRANS), b128 LDS reads,
// non-temporal b128 stores.
// WMMA intentionally absent: fp32 WMMA is 16x16x4 (9/1024 MACs for a 3x3),
// per-channel matrices cannot batch into one tile, and a full WMMA-layout
// pipeline costs ~1.3x the VALU/TRANS work plus 4x the VMEM instruction
// count -- strictly slower for an HBM-bound op.
// ---------------------------------------------------------------------------

#define PSTRIDE 48   // floats per channel in the precomputed param table (43 used)

typedef float v4f __attribute__((ext_vector_type(4)));

__device__ __forceinline__ float tanh_hw(float x) {
#if __has_builtin(__builtin_amdgcn_tanhf)
    return __builtin_amdgcn_tanhf(x);           // compiler manages TRANS hazards
#elif __has_builtin(__builtin_amdgcn_tanh_f32)
    return __builtin_amdgcn_tanh_f32(x);
#else
    float r;
    // CDNA5 native tanh; spec (§7.4): 1 independent op or V_NOP after a TRANS
    // op before its output is consumed.
    asm("v_tanh_f32 %0, %1\n\tv_nop" : "=v"(r) : "v"(x));
    return r;
#endif
}

// ---------------------------------------------------------------------------
// Prep: per-channel parameter table (runs once, 1 block, runtime noise).
//  [0..2] w0=softplus(m0)  [3..5] b0   [6..8]  t0=tanh(f0)
//  [9..17] w1              [18..20] b1 [21..23] t1
//  [24..32] w2             [33..35] b2 [36..38] t2
//  [39..41] w3             [42] b3
// ---------------------------------------------------------------------------
__global__ void cdf_prep_params(const float* __restrict__ m0, const float* __restrict__ b0,
                                const float* __restrict__ m1, const float* __restrict__ b1,
                                const float* __restrict__ m2, const float* __restrict__ b2,
                                const float* __restrict__ m3, const float* __restrict__ b3,
                                const float* __restrict__ f0, const float* __restrict__ f1,
                                const float* __restrict__ f2,
                                float* __restrict__ P, int ch) {
    int c = blockIdx.x * blockDim.x + threadIdx.x;
    if (c >= ch) return;
    float* p = P + (size_t)c * PSTRIDE;
    auto sp = [](float x) { return x > 20.0f ? x : log1pf(expf(x)); };  // softplus
#pragma unroll
    for (int j = 0; j < 3; ++j) p[0 + j]  = sp(m0[c * 3 + j]);
#pragma unroll
    for (int j = 0; j < 3; ++j) p[3 + j]  = b0[c * 3 + j];
#pragma unroll
    for (int j = 0; j < 3; ++j) p[6 + j]  = tanhf(f0[c * 3 + j]);
#pragma unroll
    for (int j = 0; j < 9; ++j) p[9 + j]  = sp(m1[c * 9 + j]);
#pragma unroll
    for (int j = 0; j < 3; ++j) p[18 + j] = b1[c * 3 + j];
#pragma unroll
    for (int j = 0; j < 3; ++j) p[21 + j] = tanhf(f1[c * 3 + j]);
#pragma unroll
    for (int j = 0; j < 9; ++j) p[24 + j] = sp(m2[c * 9 + j]);
#pragma unroll
    for (int j = 0; j < 3; ++j) p[33 + j] = b2[c * 3 + j];
#pragma unroll
    for (int j = 0; j < 3; ++j) p[36 + j] = tanhf(f2[c * 3 + j]);
#pragma unroll
    for (int j = 0; j < 3; ++j) p[39 + j] = sp(m3[c * 3 + j]);
    p[42] = b3[c];
}

struct Params {
    float w0[3], B0[3], t0[3];
    float w1[9], B1[3], t1[3];
    float w2[9], B2[3], t2[3];
    float w3[3], B3;
};

// Push 4 samples through the 4 layers, fully in registers.
__device__ __forceinline__ v4f mlp4(v4f x4, const Params& q) {
    float a[4] = {x4.x, x4.y, x4.z, x4.w};
    float h[4][3], g[4][3], z[4][3];
#pragma unroll
    for (int i = 0; i < 4; ++i)
#pragma unroll
        for (int j = 0; j < 3; ++j) {               // layer 0: 1 -> 3, gated
            float v = fmaf(q.w0[j], a[i], q.B0[j]);
            h[i][j] = fmaf(q.t0[j], tanh_hw(v), v);
        }
#pragma unroll
    for (int i = 0; i < 4; ++i)
#pragma unroll
        for (int j = 0; j < 3; ++j) {               // layer 1: 3 -> 3, gated
            float v = fmaf(q.w1[j * 3 + 0], h[i][0],
                      fmaf(q.w1[j * 3 + 1], h[i][1],
                      fmaf(q.w1[j * 3 + 2], h[i][2], q.B1[j])));
            g[i][j] = fmaf(q.t1[j], tanh_hw(v), v);
        }
#pragma unroll
    for (int i = 0; i < 4; ++i)
#pragma unroll
        for (int j = 0; j < 3; ++j) {               // layer 2: 3 -> 3, gated
            float v = fmaf(q.w2[j * 3 + 0], g[i][0],
                      fmaf(q.w2[j * 3 + 1], g[i][1],
                      fmaf(q.w2[j * 3 + 2], g[i][2], q.B2[j])));
            z[i][j] = fmaf(q.t2[j], tanh_hw(v), v);
        }
    v4f o;                                           // layer 3: 3 -> 1
    o.x = fmaf(q.w3[0], z[0][0], fmaf(q.w3[1], z[0][1], fmaf(q.w3[2], z[0][2], q.B3)));
    o.y = fmaf(q.w3[0], z[1][0], fmaf(q.w3[1], z[1][1], fmaf(q.w3[2], z[1][2], q.B3)));
    o.z = fmaf(q.w3[0], z[2][0], fmaf(q.w3[1], z[2][1], fmaf(q.w3[2], z[2][2], q.B3)));
    o.w = fmaf(q.w3[0], z[3][0], fmaf(q.w3[1], z[3][1], fmaf(q.w3[2], z[3][2], q.B3)));
    return o;
}

// ---------------------------------------------------------------------------
// Main kernel: grid = (N/4096, ch), 256 threads (8 wave32) per block.
// One clause of four async b128 copies per lane (ASYNCcnt = 4, 16 KB LDS),
// sharing one (lds, voffset) VGPR pair -- IOFFSET applies to both the LDS
// and global addresses per the ISA pseudocode. Drained with descending
// s_wait_asynccnt so tile-k compute overlaps the copies for tiles k+1..3.
// Each lane reads back exactly the LDS bytes its own async op wrote -> no
// workgroup barrier needed; waits are per-wave.
// ---------------------------------------------------------------------------
__global__ __launch_bounds__(256) void cdf_mlp_kernel(const float* __restrict__ in,
                                                      const float* __restrict__ P,
                                                      float* __restrict__ out, int N) {
    __shared__ __align__(16) float tile[4096];       // 16 KB input staging

    const int      c    = blockIdx.y;
    const unsigned tid  = threadIdx.x;
    const size_t   base = (size_t)c * (size_t)N + (size_t)blockIdx.x * 4096u;
    const float* __restrict__ gsrc = in + base;      // uniform base -> SGPR pair

    // --- clause: 4 async global->LDS b128 copies (single-use input -> NT) ----
    const unsigned lds0 = (unsigned)(size_t)tile + tid * 16u;
    const unsigned off0 = tid * 16u;
    asm volatile(
        "s_clause 0x3\n\t"
        "global_load_async_to_lds_b128 %0, %1, %2 th:TH_LOAD_NT\n\t"
        "global_load_async_to_lds_b128 %0, %1, %2 offset:4096 th:TH_LOAD_NT\n\t"
        "global_load_async_to_lds_b128 %0, %1, %2 offset:8192 th:TH_LOAD_NT\n\t"
        "global_load_async_to_lds_b128 %0, %1, %2 offset:12288 th:TH_LOAD_NT"
        :: "v"(lds0), "v"(off0), "s"(gsrc) : "memory");

    // --- overlap: pull the uniform per-channel params (43 floats) ------------
    const float* __restrict__ p = P + (size_t)c * PSTRIDE;
    Params q;
#pragma unroll
    for (int j = 0; j < 3; ++j) { q.w0[j] = p[j];      q.B0[j] = p[3 + j];  q.t0[j] = p[6 + j]; }
#pragma unroll
    for (int j = 0; j < 9; ++j) { q.w1[j] = p[9 + j];  q.w2[j] = p[24 + j]; }
#pragma unroll
    for (int j = 0; j < 3; ++j) { q.B1[j] = p[18 + j]; q.t1[j] = p[21 + j];
                                  q.B2[j] = p[33 + j]; q.t2[j] = p[36 + j];
                                  q.w3[j] = p[39 + j]; }
    q.B3 = p[42];

    // --- drain the pipeline: tile k under wait(asynccnt <= 3-k) --------------
    asm volatile("s_wait_asynccnt 0x3" ::: "memory");
    v4f x = *reinterpret_cast<const v4f*>(&tile[tid * 4u]);
    __builtin_nontemporal_store(mlp4(x, q), reinterpret_cast<v4f*>(out + base) + tid);

    asm volatile("s_wait_asynccnt 0x2" ::: "memory");
    x = *reinterpret_cast<const v4f*>(&tile[1024u + tid * 4u]);
    __builtin_nontemporal_store(mlp4(x, q), reinterpret_cast<v4f*>(out + base + 1024u) + tid);

    asm volatile("s_wait_asynccnt 0x1" ::: "memory");
    x = *reinterpret_cast<const v4f*>(&tile[2048u + tid * 4u]);
    __builtin_nontemporal_store(mlp4(x, q), reinterpret_cast<v4f*>(out + base + 2048u) + tid);

    asm volatile("s_wait_asynccnt 0x0" ::: "memory");
    x = *reinterpret_cast<const v4f*>(&tile[3072u + tid * 4u]);
    __builtin_nontemporal_store(mlp4(x, q), reinterpret_cast<v4f*>(out + base + 3072u) + tid);
}

// ---------------------------------------------------------------------------
// Harness entry. d_in dict order:
//  0 inputs, 1 m0, 2 b0, 3 m1, 4 b1, 5 m2, 6 b2, 7 m3, 8 b3, 9 f0, 10 f1,
//  11 f2, 12 stop_gradient
// ---------------------------------------------------------------------------
extern "C" void kernel_launch(void* const* d_in, const int* in_sizes, int n_in,
                              void* d_out, int out_size, void* d_ws, size_t ws_size,
                              hipStream_t stream) {
    const float* inputs = (const float*)d_in[0];
    const float* m0 = (const float*)d_in[1];
    const float* b0 = (const float*)d_in[2];
    const float* m1 = (const float*)d_in[3];
    const float* b1 = (const float*)d_in[4];
    const float* m2 = (const float*)d_in[5];
    const float* b2 = (const float*)d_in[6];
    const float* m3 = (const float*)d_in[7];
    const float* b3 = (const float*)d_in[8];
    const float* f0 = (const float*)d_in[9];
    const float* f1 = (const float*)d_in[10];
    const float* f2 = (const float*)d_in[11];

    const int ch = in_sizes[1] / 3;        // m0 is [ch,3,1]
    const int N  = in_sizes[0] / ch;       // inputs is [ch,1,N]

    float* P = (float*)d_ws;               // ch * PSTRIDE floats (48 KB for ch=256)

    cdf_prep_params<<<dim3((ch + 255) / 256), dim3(256), 0, stream>>>(
        m0, b0, m1, b1, m2, b2, m3, b3, f0, f1, f2, P, ch);

    dim3 grid(N / 4096, ch);
    cdf_mlp_kernel<<<grid, dim3(256), 0, stream>>>(inputs, P, (float*)d_out, N);
}